// GAT_7327214207309
// MI455X (gfx1250) — compile-verified
//
#include <hip/hip_runtime.h>
#include <hip/hip_bf16.h>
#include <math.h>

// ---------------------------------------------------------------------------
// GAT 3-layer forward for MI455X (gfx1250, wave32).
//  - GEMMs: V_WMMA_F32_16X16X4_F32, A tiles staged in LDS by the Tensor Data
//    Mover (tensor_load_to_lds + s_wait_tensorcnt), double-buffered.
//  - Edge phase: ordered-int atomic max (exact segment softmax), f32 atomics.
// ---------------------------------------------------------------------------

typedef __attribute__((ext_vector_type(2))) float v2f;
typedef __attribute__((ext_vector_type(8))) float v8f;
typedef __attribute__((ext_vector_type(4))) unsigned int u32x4;
typedef __attribute__((ext_vector_type(8))) int i32x8;
typedef __attribute__((ext_vector_type(4))) int i32x4;

#define NEG_SLOPE 0.2f
#define GEMM_KC   64      // K-chunk staged in LDS per step
#define GEMM_WPB  4       // waves per block (128 threads)

#if __has_builtin(__builtin_amdgcn_tensor_load_to_lds) && \
    __has_builtin(__builtin_amdgcn_s_wait_tensorcnt)
#define USE_TDM 1
#else
#define USE_TDM 0
#endif

// ---- ordered-int encoding for float atomic max ----------------------------
__device__ __forceinline__ int f2ord(float f) {
    int i = __float_as_int(f);
    return (i < 0) ? (i ^ 0x7FFFFFFF) : i;
}
__device__ __forceinline__ float ord2f(int i) {
    return __int_as_float((i < 0) ? (i ^ 0x7FFFFFFF) : i);
}
#define ORD_NEG_INF ((int)0x807FFFFF)   // encoded -inf

#if USE_TDM
// ---------------------------------------------------------------------------
// TDM: DMA a 16 x GEMM_KC f32 tile (row-major, zero-filled OOB) into LDS.
// D# built per CDNA5 ISA:
//   group0: [1:0]=count=1, [63:32]=lds_addr, [120:64]=global_addr, [127:126]=2
//   group1: data_size=4B, tensor_dim0=remK, tensor_dim1=remRows,
//           tile_dim0=KC, tile_dim1=16, tensor_dim0_stride=Kdim
// 6-arg builtin form (clang-23 / therock-10.0 headers).
// ---------------------------------------------------------------------------
__device__ __forceinline__ void tdm_load_tile(const float* gsrc, unsigned ldsOff,
                                              unsigned remK, unsigned remRows,
                                              unsigned long long strideElems) {
    unsigned long long ga = (unsigned long long)(__UINTPTR_TYPE__)gsrc;
    u32x4 g0;
    g0[0] = 1u;                                           // count=1, user mode
    g0[1] = ldsOff;                                       // LDS byte address
    g0[2] = (unsigned)ga;                                 // global_addr[31:0]
    g0[3] = (unsigned)((ga >> 32) & 0x1FFFFFFu) | (2u << 30);  // addr[56:32]|type=2
    unsigned td0 = remK, td1 = remRows;
    unsigned tile0 = GEMM_KC, tile1 = 16u;
    i32x8 g1;
    g1[0] = (int)(2u << 16);                              // data_size = 4 bytes
    g1[1] = (int)((td0 & 0xFFFFu) << 16);                 // tensor_dim0 lo
    g1[2] = (int)((td0 >> 16) | ((td1 & 0xFFFFu) << 16)); // dim0 hi | dim1 lo
    g1[3] = (int)((td1 >> 16) | (tile0 << 16));           // dim1 hi | tile_dim0
    g1[4] = (int)tile1;                                   // tile_dim1 (tile_dim2=0)
    g1[5] = (int)(strideElems & 0xFFFFFFFFull);           // dim0_stride lo
    g1[6] = (int)((strideElems >> 32) & 0xFFFFull);       // dim0_stride hi
    g1[7] = 0;                                            // dim1_stride (unused)
    i32x4 z4 = {0, 0, 0, 0};                              // groups 2/3 unused (2D)
    i32x8 z8 = {0, 0, 0, 0, 0, 0, 0, 0};
    __builtin_amdgcn_tensor_load_to_lds(g0, g1, z4, z4, z8, 0);
}
#endif

// ---------------------------------------------------------------------------
// WMMA f32 GEMM: C[Nrows,Ncols] = A[Nrows,Kdim] @ B[Kdim,Ncols]
// One block = GEMM_WPB waves = one 16-row strip; wave w owns column tile w.
// A chunk (16 x 64 f32) staged in LDS (TDM double-buffered); B from global/L2.
// Fragment layouts per ISA 7.12.2 (A: K pairs split across lane halves;
// B/C: row striped across lanes).  No predicated loads: OOB handled by
// TDM zero-fill (A) and address clamps (B).
// ---------------------------------------------------------------------------
__global__ __launch_bounds__(GEMM_WPB * 32)
void gat_gemm_wmma(const float* __restrict__ A, const float* __restrict__ B,
                   float* __restrict__ C, int Nrows, int Kdim, int Ncols) {
    __shared__ float sA[2][16 * GEMM_KC];

    const int lane  = threadIdx.x & 31;
    const int wib   = threadIdx.x >> 5;
    const int half  = lane >> 4;          // 0|1
    const int half2 = half * 2;
    const int l16   = lane & 15;

    const int colTiles = (Ncols + 15) >> 4;
    const int row0  = blockIdx.x << 4;
    const int tileC = (wib < colTiles) ? wib : (colTiles - 1);
    const bool isOwner = (wib < colTiles);
    const int col0  = tileC << 4;

    const int bCol  = col0 + l16;
    const bool bOk  = (bCol < Ncols);
    const int colC  = bOk ? bCol : (Ncols - 1);           // clamp, no branch

    const int nChunks = (Kdim + GEMM_KC - 1) / GEMM_KC;

#if USE_TDM
    if (wib == 0) {
        tdm_load_tile(A + (size_t)row0 * Kdim, (unsigned)(__UINTPTR_TYPE__)&sA[0][0],
                      (unsigned)Kdim, (unsigned)(Nrows - row0),
                      (unsigned long long)Kdim);
    }
#endif

    v8f acc = {};
    for (int ic = 0; ic < nChunks; ++ic) {
        const int k0 = ic * GEMM_KC;
#if USE_TDM
        if (wib == 0) {
            if (ic + 1 < nChunks) {
                const int kn = k0 + GEMM_KC;
                tdm_load_tile(A + (size_t)row0 * Kdim + kn,
                              (unsigned)(__UINTPTR_TYPE__)&sA[(ic + 1) & 1][0],
                              (unsigned)(Kdim - kn), (unsigned)(Nrows - row0),
                              (unsigned long long)Kdim);
                __builtin_amdgcn_s_wait_tensorcnt(1);   // chunk ic complete
            } else {
                __builtin_amdgcn_s_wait_tensorcnt(0);
            }
        }
        __syncthreads();                                // publish chunk ic
#else
        __syncthreads();
        for (int idx = threadIdx.x; idx < 16 * GEMM_KC; idx += GEMM_WPB * 32) {
            int r = idx >> 6, c = idx & (GEMM_KC - 1);
            int gr = row0 + r, gc = k0 + c;
            sA[ic & 1][idx] = (gr < Nrows && gc < Kdim)
                                  ? A[(size_t)gr * Kdim + gc] : 0.f;
        }
        __syncthreads();
#endif
        const float* __restrict__ sAc = &sA[ic & 1][0];
#pragma unroll
        for (int kk = 0; kk < GEMM_KC; kk += 4) {
            const int kg  = k0 + kk + half2;
            const int kc0 = (kg     < Kdim) ? kg       : (Kdim - 1);  // clamp
            const int kc1 = (kg + 1 < Kdim) ? (kg + 1) : (Kdim - 1);  // clamp
            const float2 af = *(const float2*)(sAc + l16 * GEMM_KC + kk + half2);
            v2f a, b;
            a.x = af.x;                       // zero-padded by TDM for OOB K
            a.y = af.y;
            b.x = B[(size_t)kc0 * Ncols + colC];   // garbage * 0 = 0 on tails
            b.y = B[(size_t)kc1 * Ncols + colC];
            acc = __builtin_amdgcn_wmma_f32_16x16x4_f32(
                false, a, false, b, (short)0, acc, false, false);
        }
        __syncthreads();                                // done reading chunk ic
    }

#pragma unroll
    for (int r = 0; r < 8; ++r) {
        int row = row0 + r + half * 8;
        if (isOwner && bOk && row < Nrows) C[(size_t)row * Ncols + bCol] = acc[r];
    }
}

// ---------------------------------------------------------------------------
// alpha_src / alpha_dst : per (node, head) dot of h[n,h,:] with a_src/a_dst
// ---------------------------------------------------------------------------
__global__ void gat_alpha(const float* __restrict__ h,
                          const float* __restrict__ a_src,
                          const float* __restrict__ a_dst,
                          float* __restrict__ as, float* __restrict__ ad,
                          int N, int H, int O) {
    int t = blockIdx.x * blockDim.x + threadIdx.x;
    if (t >= N * H) return;
    int n = t / H, hh = t - n * H;
    const float* hp = h + (size_t)n * H * O + (size_t)hh * O;
    const float* sp = a_src + hh * O;
    const float* dp = a_dst + hh * O;
    float s1 = 0.f, s2 = 0.f;
    for (int o = 0; o < O; ++o) { float v = hp[o]; s1 += v * sp[o]; s2 += v * dp[o]; }
    as[t] = s1; ad[t] = s2;
}

// init: m = ord(-inf), ssum = 0, agg = 0
__global__ void gat_init(int* __restrict__ m, float* __restrict__ ss,
                         float* __restrict__ agg, int NH, int NHO) {
    int t = blockIdx.x * blockDim.x + threadIdx.x;
    if (t < NH) { m[t] = ORD_NEG_INF; ss[t] = 0.f; }
    if (t < NHO) agg[t] = 0.f;
}

__device__ __forceinline__ void edge_ends(const int* __restrict__ src,
                                          const int* __restrict__ dst,
                                          int e, int E, int& s, int& d) {
    if (e < E) { s = src[e]; d = dst[e]; }
    else       { s = e - E;  d = e - E;  }  // self-loops appended
}

// pass 1: per-destination max of leaky_relu(alpha_src[s]+alpha_dst[d])
__global__ void gat_edge_max(const int* __restrict__ src, const int* __restrict__ dst,
                             int E, long long EH, int H,
                             const float* __restrict__ as, const float* __restrict__ ad,
                             int* __restrict__ m) {
    long long t = (long long)blockIdx.x * blockDim.x + threadIdx.x;
    if (t >= EH) return;
    int e = (int)(t / H), hh = (int)(t - (long long)e * H);
    int s, d; edge_ends(src, dst, e, E, s, d);
    float x = as[(size_t)s * H + hh] + ad[(size_t)d * H + hh];
    x = (x > 0.f) ? x : NEG_SLOPE * x;
    atomicMax(&m[(size_t)d * H + hh], f2ord(x));
}

// pass 2: ex = exp(e - m[dst]); store per-edge weight; segment-sum into ss
__global__ void gat_edge_expsum(const int* __restrict__ src, const int* __restrict__ dst,
                                int E, long long EH, int H,
                                const float* __restrict__ as, const float* __restrict__ ad,
                                const int* __restrict__ m,
                                float* __restrict__ ae, float* __restrict__ ss) {
    long long t = (long long)blockIdx.x * blockDim.x + threadIdx.x;
    if (t >= EH) return;
    int e = (int)(t / H), hh = (int)(t - (long long)e * H);
    int s, d; edge_ends(src, dst, e, E, s, d);
    float x = as[(size_t)s * H + hh] + ad[(size_t)d * H + hh];
    x = (x > 0.f) ? x : NEG_SLOPE * x;
    float ex = expf(x - ord2f(m[(size_t)d * H + hh]));
    ae[t] = ex;
    atomicAdd(&ss[(size_t)d * H + hh], ex);
}

// pass 3: agg[dst,h,o] += h[src,h,o] * ae[e,h] / (ss[dst,h] + 1e-16)
__global__ void gat_scatter(const int* __restrict__ src, const int* __restrict__ dst,
                            int E, long long EHO, int H, int O,
                            const float* __restrict__ h,
                            const float* __restrict__ ae,
                            const float* __restrict__ ss,
                            float* __restrict__ agg) {
    long long t = (long long)blockIdx.x * blockDim.x + threadIdx.x;
    if (t >= EHO) return;
    int HO = H * O;
    int e  = (int)(t / HO);
    int r  = (int)(t - (long long)e * HO);
    int hh = r / O;
    int s, d; edge_ends(src, dst, e, E, s, d);
    float w = ae[(size_t)e * H + hh] / (ss[(size_t)d * H + hh] + 1e-16f);
    atomicAdd(&agg[(size_t)d * HO + r], h[(size_t)s * HO + r] * w);
}

// concat layers: out = elu(agg + bias)
__global__ void gat_bias_elu(const float* __restrict__ agg, const float* __restrict__ bias,
                             float* __restrict__ out, int N, int HO) {
    int t = blockIdx.x * blockDim.x + threadIdx.x;
    if (t >= N * HO) return;
    int r = t % HO;
    float v = agg[t] + bias[r];
    out[t] = (v > 0.f) ? v : (expf(v) - 1.f);
}

// final layer: mean over heads + bias, elu, log_softmax over O classes
__global__ void gat_final(const float* __restrict__ agg, const float* __restrict__ bias,
                          float* __restrict__ out, int N, int H, int O) {
    int n = blockIdx.x * blockDim.x + threadIdx.x;
    if (n >= N) return;
    float v[16];  // O <= 16
    float inv = 1.f / (float)H;
    for (int c = 0; c < O; ++c) {
        float s = 0.f;
        for (int hh = 0; hh < H; ++hh) s += agg[(size_t)n * H * O + hh * O + c];
        s = s * inv + bias[c];
        v[c] = (s > 0.f) ? s : (expf(s) - 1.f);
    }
    float mx = v[0];
    for (int c = 1; c < O; ++c) mx = fmaxf(mx, v[c]);
    float se = 0.f;
    for (int c = 0; c < O; ++c) se += expf(v[c] - mx);
    float lse = mx + logf(se);
    for (int c = 0; c < O; ++c) out[(size_t)n * O + c] = v[c] - lse;
}

// ---------------------------------------------------------------------------
// host-side driver
// ---------------------------------------------------------------------------
static void run_gat_layer(const float* xin, int Fin,
                          const float* W, const float* a_s, const float* a_d,
                          const float* bias, int H, int O, int final_mode,
                          const int* src, const int* dst, int E, int Etot, int N,
                          float* h, float* agg, float* as, float* ad, int* m,
                          float* ss, float* ae, float* out, hipStream_t stream) {
    const int HO = H * O;
    // GEMM: h = xin @ W  (one block per 16-row strip, 4 waves = 4 col tiles)
    {
        int rowTiles = (N + 15) / 16;
        gat_gemm_wmma<<<rowTiles, GEMM_WPB * 32, 0, stream>>>(xin, W, h, N, Fin, HO);
    }
    int NH  = N * H;
    int NHO = N * HO;
    gat_alpha<<<(NH + 255) / 256, 256, 0, stream>>>(h, a_s, a_d, as, ad, N, H, O);
    gat_init<<<(NHO + 255) / 256, 256, 0, stream>>>(m, ss, agg, NH, NHO);

    long long EH = (long long)Etot * H;
    int ehBlocks = (int)((EH + 255) / 256);
    gat_edge_max<<<ehBlocks, 256, 0, stream>>>(src, dst, E, EH, H, as, ad, m);
    gat_edge_expsum<<<ehBlocks, 256, 0, stream>>>(src, dst, E, EH, H, as, ad, m, ae, ss);

    long long EHO = (long long)Etot * HO;
    int ehoBlocks = (int)((EHO + 255) / 256);
    gat_scatter<<<ehoBlocks, 256, 0, stream>>>(src, dst, E, EHO, H, O, h, ae, ss, agg);

    if (!final_mode)
        gat_bias_elu<<<(NHO + 255) / 256, 256, 0, stream>>>(agg, bias, out, N, HO);
    else
        gat_final<<<(N + 255) / 256, 256, 0, stream>>>(agg, bias, out, N, H, O);
}

extern "C" void kernel_launch(void* const* d_in, const int* in_sizes, int n_in,
                              void* d_out, int out_size, void* d_ws, size_t ws_size,
                              hipStream_t stream) {
    const float* x   = (const float*)d_in[0];
    const int*   ei  = (const int*)d_in[1];      // [2, E] int32
    const float* W1  = (const float*)d_in[2];
    const float* a1s = (const float*)d_in[3];
    const float* a1d = (const float*)d_in[4];
    const float* b1  = (const float*)d_in[5];
    const float* W2  = (const float*)d_in[6];
    const float* a2s = (const float*)d_in[7];
    const float* a2d = (const float*)d_in[8];
    const float* b2  = (const float*)d_in[9];
    const float* W3  = (const float*)d_in[10];
    const float* a3s = (const float*)d_in[11];
    const float* a3d = (const float*)d_in[12];
    const float* b3  = (const float*)d_in[13];

    const int F_IN = 1433;
    const int N    = in_sizes[0] / F_IN;
    const int E    = in_sizes[1] / 2;
    const int Etot = E + N;                      // self-loops appended virtually
    const int* src = ei;
    const int* dst = ei + E;

    // carve workspace
    char*  ws  = (char*)d_ws;
    size_t off = 0;
    auto carve = [&](size_t bytes) -> void* {
        off = (off + 255) & ~(size_t)255;
        void* p = ws + off;
        off += bytes;
        return p;
    };
    float* hA  = (float*)carve((size_t)N * 64 * sizeof(float));  // layer input (post-ELU)
    float* hB  = (float*)carve((size_t)N * 64 * sizeof(float));  // GEMM output h
    float* agg = (float*)carve((size_t)N * 64 * sizeof(float));  // scatter accumulator
    float* as_ = (float*)carve((size_t)N * 6 * sizeof(float));
    float* ad_ = (float*)carve((size_t)N * 6 * sizeof(float));
    int*   m   = (int*)  carve((size_t)N * 6 * sizeof(int));
    float* ss  = (float*)carve((size_t)N * 6 * sizeof(float));
    float* ae  = (float*)carve((size_t)Etot * 6 * sizeof(float));
    (void)ws_size; (void)n_in; (void)out_size;

    // layer 1: 1433 -> 4x16 concat, ELU   (x -> hB -> agg -> hA)
    run_gat_layer(x, F_IN, W1, a1s, a1d, b1, 4, 16, 0,
                  src, dst, E, Etot, N, hB, agg, as_, ad_, m, ss, ae, hA, stream);
    // layer 2: 64 -> 4x16 concat, ELU     (hA -> hB -> agg -> hA)
    run_gat_layer(hA, 64, W2, a2s, a2d, b2, 4, 16, 0,
                  src, dst, E, Etot, N, hB, agg, as_, ad_, m, ss, ae, hA, stream);
    // layer 3: 64 -> 6x7 mean, ELU, log_softmax -> d_out
    run_gat_layer(hA, 64, W3, a3s, a3d, b3, 6, 7, 1,
                  src, dst, E, Etot, N, hB, agg, as_, ad_, m, ss, ae,
                  (float*)d_out, stream);
}